// ARCModel_43903155699729
// MI455X (gfx1250) — compile-verified
//
#include <hip/hip_runtime.h>
#include <math.h>

typedef __attribute__((ext_vector_type(2))) float v2f;
typedef __attribute__((ext_vector_type(8))) float v8f;

#define NUM_HEADS 4
#define HEAD_DIM  64
#define RHO       0.1f
#define LSEQ      225
#define LP        240   // padded L (15*16)

#define WMMA_F32(a, b, c) \
  __builtin_amdgcn_wmma_f32_16x16x4_f32(false, (a), false, (b), (short)0, (c), false, false)

// ---------------------------------------------------------------------------
// WMMA f32 16x16x4 fragment layout (wave32):
//  A (16x4): lanes 0-15 row=lane, VGPR0/1 = K,K+1 ; lanes 16-31 = K+2,K+3
//  B (4x16): mirrored (cols across lanes, K pair selected by lane half)
//  C/D     : VGPR r -> row r (lanes 0-15) / row r+8 (lanes 16-31), col = lane&15
// One wave computes a 32x64 block: 2 A fragments x 4 B fragments = 8 acc
// chains. K-loop is register double-buffered so next-step fragment loads
// overlap current-step WMMAs (partial loadcnt waits instead of wait-0).
// Loads are clamped (no predication); stores are bounds-checked; K % 4 == 0.
// ---------------------------------------------------------------------------
__device__ __forceinline__ void store_tile_f32(float* __restrict__ C, int ldc,
                                               int row0, int col0, int M, int N,
                                               v8f acc, float alpha) {
  const int lane = threadIdx.x & 31;
  const int half = lane >> 4;
  const int l16  = lane & 15;
  const int col  = col0 + l16;
  if (col < N) {
#pragma unroll
    for (int r = 0; r < 8; ++r) {
      const int row = row0 + r + 8 * half;
      if (row < M) C[(size_t)row * ldc + col] = alpha * acc[r];
    }
  }
}

template <bool TB>
__device__ __forceinline__ v2f load_b_frag(const float* __restrict__ Bp, int k0, int ldb) {
  v2f b;
  if (TB) {
    b = *(const v2f*)(Bp + k0);
  } else {
    b.x = Bp[(size_t)k0 * ldb];
    b.y = Bp[(size_t)(k0 + 1) * ldb];
  }
  return b;
}

// Generic batched GEMM: C = alpha * A * op(B).
// blockIdx.z decomposed as (b = z / nh, h = z % nh); per-operand (b,h) strides.
template <bool TB>
__global__ void gemm_wmma(const float* __restrict__ A, const float* __restrict__ B,
                          float* __restrict__ C, int M, int N, int K,
                          int lda, int ldb, int ldc, int nh,
                          long long sAb, long long sAh,
                          long long sBb, long long sBh,
                          long long sCb, long long sCh, float alpha) {
  const int bz = blockIdx.z;
  const int b  = bz / nh;
  const int h  = bz % nh;
  const float* Ab = A + (size_t)b * sAb + (size_t)h * sAh;
  const float* Bb = B + (size_t)b * sBb + (size_t)h * sBh;
  float*       Cb = C + (size_t)b * sCb + (size_t)h * sCh;

  const int lane = threadIdx.x & 31;
  const int half = lane >> 4;
  const int l16  = lane & 15;
  const int koff = 2 * half;

  const int row0 = blockIdx.x * 32;
  int ar0 = row0 + l16;      if (ar0 > M - 1) ar0 = M - 1;   // clamp: no predication
  int ar1 = row0 + 16 + l16; if (ar1 > M - 1) ar1 = M - 1;
  const float* Ap0 = Ab + (size_t)ar0 * lda + koff;
  const float* Ap1 = Ab + (size_t)ar1 * lda + koff;

  const int col0 = blockIdx.y * 64;
  const float* Bp0; const float* Bp1; const float* Bp2; const float* Bp3;
  {
    int c0 = col0      + l16; if (c0 > N - 1) c0 = N - 1;
    int c1 = col0 + 16 + l16; if (c1 > N - 1) c1 = N - 1;
    int c2 = col0 + 32 + l16; if (c2 > N - 1) c2 = N - 1;
    int c3 = col0 + 48 + l16; if (c3 > N - 1) c3 = N - 1;
    if (TB) {
      Bp0 = Bb + (size_t)c0 * ldb + koff;
      Bp1 = Bb + (size_t)c1 * ldb + koff;
      Bp2 = Bb + (size_t)c2 * ldb + koff;
      Bp3 = Bb + (size_t)c3 * ldb + koff;
    } else {
      Bp0 = Bb + (size_t)koff * ldb + c0;
      Bp1 = Bb + (size_t)koff * ldb + c1;
      Bp2 = Bb + (size_t)koff * ldb + c2;
      Bp3 = Bb + (size_t)koff * ldb + c3;
    }
  }

  v8f acc00 = {}, acc01 = {}, acc02 = {}, acc03 = {};
  v8f acc10 = {}, acc11 = {}, acc12 = {}, acc13 = {};

  // Prologue: fragments for k0 = 0
  v2f a0c = *(const v2f*)(Ap0);
  v2f a1c = *(const v2f*)(Ap1);
  v2f b0c = load_b_frag<TB>(Bp0, 0, ldb);
  v2f b1c = load_b_frag<TB>(Bp1, 0, ldb);
  v2f b2c = load_b_frag<TB>(Bp2, 0, ldb);
  v2f b3c = load_b_frag<TB>(Bp3, 0, ldb);

  for (int k0 = 4; k0 < K; k0 += 4) {
    // Prefetch next k-step (independent of the WMMAs below)
    v2f a0n = *(const v2f*)(Ap0 + k0);
    v2f a1n = *(const v2f*)(Ap1 + k0);
    v2f b0n = load_b_frag<TB>(Bp0, k0, ldb);
    v2f b1n = load_b_frag<TB>(Bp1, k0, ldb);
    v2f b2n = load_b_frag<TB>(Bp2, k0, ldb);
    v2f b3n = load_b_frag<TB>(Bp3, k0, ldb);

    acc00 = WMMA_F32(a0c, b0c, acc00);
    acc01 = WMMA_F32(a0c, b1c, acc01);
    acc02 = WMMA_F32(a0c, b2c, acc02);
    acc03 = WMMA_F32(a0c, b3c, acc03);
    acc10 = WMMA_F32(a1c, b0c, acc10);
    acc11 = WMMA_F32(a1c, b1c, acc11);
    acc12 = WMMA_F32(a1c, b2c, acc12);
    acc13 = WMMA_F32(a1c, b3c, acc13);

    a0c = a0n; a1c = a1n;
    b0c = b0n; b1c = b1n; b2c = b2n; b3c = b3n;
  }
  // Epilogue: last k-step
  acc00 = WMMA_F32(a0c, b0c, acc00);
  acc01 = WMMA_F32(a0c, b1c, acc01);
  acc02 = WMMA_F32(a0c, b2c, acc02);
  acc03 = WMMA_F32(a0c, b3c, acc03);
  acc10 = WMMA_F32(a1c, b0c, acc10);
  acc11 = WMMA_F32(a1c, b1c, acc11);
  acc12 = WMMA_F32(a1c, b2c, acc12);
  acc13 = WMMA_F32(a1c, b3c, acc13);

  store_tile_f32(Cb, ldc, row0,      col0,      M, N, acc00, alpha);
  store_tile_f32(Cb, ldc, row0,      col0 + 16, M, N, acc01, alpha);
  store_tile_f32(Cb, ldc, row0,      col0 + 32, M, N, acc02, alpha);
  store_tile_f32(Cb, ldc, row0,      col0 + 48, M, N, acc03, alpha);
  store_tile_f32(Cb, ldc, row0 + 16, col0,      M, N, acc10, alpha);
  store_tile_f32(Cb, ldc, row0 + 16, col0 + 16, M, N, acc11, alpha);
  store_tile_f32(Cb, ldc, row0 + 16, col0 + 32, M, N, acc12, alpha);
  store_tile_f32(Cb, ldc, row0 + 16, col0 + 48, M, N, acc13, alpha);
}

// Zero-fill workspace (float4 stores); n4 = count of float4 elements.
__global__ void zero_kernel(float4* __restrict__ p, int n4) {
  const int i = blockIdx.x * blockDim.x + threadIdx.x;
  if (i < n4) p[i] = make_float4(0.f, 0.f, 0.f, 0.f);
}

// ---------------------------------------------------------------------------
// Adjacency build on padded [LP x LP] S (ld = LP):
// A[i,k] = (1/L) * sum_j th(S[i,j]*S[k,j]/d), th(u)=u*(u>rho), A[i,i]=0;
// writes A_tilde = A + I and deg[i] = rowsum. Block per (row i, bh).
// ---------------------------------------------------------------------------
__global__ void adj_kernel(const float* __restrict__ S, float* __restrict__ At,
                           float* __restrict__ deg) {
  const int L  = LSEQ;
  const int i  = blockIdx.x;
  const int bh = blockIdx.y;
  const float* Sb = S + (size_t)bh * LP * LP;
  float* Atb      = At + (size_t)bh * LP * LP;
  __shared__ float srow[LSEQ];
  __shared__ float red[256];
  const int t = threadIdx.x;
  const float inv_d = 1.0f / (float)HEAD_DIM;
  const float inv_L = 1.0f / (float)L;
  if (t < L) srow[t] = Sb[(size_t)i * LP + t];
  __syncthreads();
  float rowsum = 0.0f;
  for (int k = t; k < L; k += blockDim.x) {
    float a;
    if (k == i) {
      a = 1.0f;  // A[i,i]=0, plus identity
    } else {
      const float* Sk = Sb + (size_t)k * LP;
      float acc = 0.0f;
#pragma unroll 4
      for (int j = 0; j < L; ++j) {
        const float u = srow[j] * Sk[j] * inv_d;
        acc += (u > RHO) ? u : 0.0f;
      }
      a = acc * inv_L;
    }
    Atb[(size_t)i * LP + k] = a;
    rowsum += a;
  }
  red[t] = rowsum;
  __syncthreads();
  for (int s = blockDim.x >> 1; s > 0; s >>= 1) {
    if (t < s) red[t] += red[t + s];
    __syncthreads();
  }
  if (t == 0) deg[(size_t)bh * LSEQ + i] = red[0];
}

// A_hat[i,k] = rsqrt(clip(deg_i)) * A_tilde[i,k] * rsqrt(clip(deg_k)) (in place)
__global__ void norm_kernel(float* __restrict__ At, const float* __restrict__ deg,
                            int total) {
  const int idx = blockIdx.x * blockDim.x + threadIdx.x;
  if (idx >= total) return;
  const int L  = LSEQ;
  const int LL = L * L;
  const int bh = idx / LL;
  const int r  = idx % LL;
  const int i  = r / L;
  const int k  = r % L;
  const float di = fmaxf(deg[(size_t)bh * L + i], 1e-6f);
  const float dk = fmaxf(deg[(size_t)bh * L + k], 1e-6f);
  float* p = At + (size_t)bh * LP * LP + (size_t)i * LP + k;
  *p = *p * rsqrtf(di) * rsqrtf(dk);
}

// Row softmax over L valid entries of a padded LP row, in place.
__global__ void softmax_kernel(float* __restrict__ Sj) {
  const int L = LSEQ;
  __shared__ float red[256];
  const int t = threadIdx.x;
  float* row = Sj + (size_t)blockIdx.y * LP * LP + (size_t)blockIdx.x * LP;
  float m = -INFINITY;
  for (int j = t; j < L; j += blockDim.x) m = fmaxf(m, row[j]);
  red[t] = m;
  __syncthreads();
  for (int s = blockDim.x >> 1; s > 0; s >>= 1) {
    if (t < s) red[t] = fmaxf(red[t], red[t + s]);
    __syncthreads();
  }
  m = red[0];
  __syncthreads();
  float sum = 0.0f;
  for (int j = t; j < L; j += blockDim.x) {
    const float e = __expf(row[j] - m);
    row[j] = e;
    sum += e;
  }
  red[t] = sum;
  __syncthreads();
  for (int s = blockDim.x >> 1; s > 0; s >>= 1) {
    if (t < s) red[t] += red[t + s];
    __syncthreads();
  }
  const float inv = 1.0f / red[0];
  for (int j = t; j < L; j += blockDim.x) row[j] *= inv;
}

extern "C" void kernel_launch(void* const* d_in, const int* in_sizes, int n_in,
                              void* d_out, int out_size, void* d_ws, size_t ws_size,
                              hipStream_t stream) {
  (void)in_sizes; (void)n_in; (void)out_size; (void)ws_size;
  const int B = 2, L = LSEQ, D = 256, H = NUM_HEADS, dh = HEAD_DIM, Hd = H * HEAD_DIM;
  const int BH = B * H;  // 8
  const long long LPP = (long long)LP * LP;

  const float* x  = (const float*)d_in[0];
  const float* Wq = (const float*)d_in[1];
  const float* Wk = (const float*)d_in[2];
  const float* Wv = (const float*)d_in[3];
  const float* Wo = (const float*)d_in[4];
  float* out = (float*)d_out;

  // Workspace layout (floats), everything padded to multiples of 16:
  float* ws = (float*)d_ws;
  float* q   = ws;                        // [B, LP, Hd]  2*240*256 = 122880
  float* k   = q   + (size_t)B * LP * Hd; // [B, LP, Hd]
  float* v   = k   + (size_t)B * LP * Hd; // [B, LP, Hd]
  float* oh  = v   + (size_t)B * LP * Hd; // [B, LP, Hd]
  float* S   = oh  + (size_t)B * LP * Hd; // [BH, LP, LP] (later S_jump/attn)
  float* At  = S   + (size_t)BH * LPP;    // [BH, LP, LP] (A_tilde -> A_hat)
  float* M1  = At  + (size_t)BH * LPP;    // [BH, LP, LP]
  float* deg = M1  + (size_t)BH * LPP;    // [BH, L]
  const size_t total_f = (size_t)(4 * B * LP * Hd) + (size_t)(3 * BH) * LPP + 2048;

  // Zero the workspace so all padding regions are exact zeros.
  {
    const int n4 = (int)(total_f / 4);
    zero_kernel<<<(n4 + 255) / 256, 256, 0, stream>>>((float4*)ws, n4);
  }

  const dim3 wb(32);  // one wave per block: EXEC all-ones for WMMA
  const int rt = (L + 31) / 32;  // 8 row tiles of 32

  // Projections per batch: q/k/v[b] = x[b] @ W^T   [225,256] x [256,256]^T
  gemm_wmma<true><<<dim3(rt, 4, B), wb, 0, stream>>>(
      x, Wq, q, L, Hd, D, D, D, Hd, 1,
      (long long)L * D, 0, 0, 0, (long long)LP * Hd, 0, 1.0f);
  gemm_wmma<true><<<dim3(rt, 4, B), wb, 0, stream>>>(
      x, Wk, k, L, Hd, D, D, D, Hd, 1,
      (long long)L * D, 0, 0, 0, (long long)LP * Hd, 0, 1.0f);
  gemm_wmma<true><<<dim3(rt, 4, B), wb, 0, stream>>>(
      x, Wv, v, L, Hd, D, D, D, Hd, 1,
      (long long)L * D, 0, 0, 0, (long long)LP * Hd, 0, 1.0f);

  // S = Q K^T per (b,h): [225,64] x [225,64]^T ; N=225 -> 4 col strips of 64
  gemm_wmma<true><<<dim3(rt, 4, BH), wb, 0, stream>>>(
      q, k, S, L, L, dh, Hd, Hd, LP, H,
      (long long)LP * Hd, dh, (long long)LP * Hd, dh, (long long)H * LPP, LPP, 1.0f);

  // Thresholded adjacency + degree; symmetric normalization in place
  adj_kernel<<<dim3(L, BH), 256, 0, stream>>>(S, At, deg);
  {
    const int total = BH * L * L;
    norm_kernel<<<(total + 255) / 256, 256, 0, stream>>>(At, deg, total);
  }

  // Jump: M1 = A_hat * S ; S_jump = (M1 * A_hat^T)/sqrt(d) (into S buffer).
  // K = LP = 240 (padding rows/cols are exact zeros).
  gemm_wmma<false><<<dim3(rt, 4, BH), wb, 0, stream>>>(
      At, S, M1, L, L, LP, LP, LP, LP, 1, LPP, 0, LPP, 0, LPP, 0, 1.0f);
  gemm_wmma<true><<<dim3(rt, 4, BH), wb, 0, stream>>>(
      M1, At, S, L, L, LP, LP, LP, LP, 1, LPP, 0, LPP, 0, LPP, 0, 0.125f);

  // Row softmax (in place on S)
  softmax_kernel<<<dim3(L, BH), 256, 0, stream>>>(S);

  // out_head = attn @ V per (b,h): [225,240(zero-padded)] x [240,64]
  gemm_wmma<false><<<dim3(rt, 1, BH), wb, 0, stream>>>(
      S, v, oh, L, dh, LP, LP, Hd, Hd, H,
      (long long)H * LPP, LPP, (long long)LP * Hd, dh, (long long)LP * Hd, dh, 1.0f);

  // Final per batch: out[b] = oh[b] @ Wo^T  [225,256] x [256,256]^T -> [225,256]
  gemm_wmma<true><<<dim3(rt, 4, B), wb, 0, stream>>>(
      oh, Wo, out, L, D, Hd, Hd, Hd, D, 1,
      (long long)LP * Hd, 0, 0, 0, (long long)L * D, 0, 1.0f);
}